// HeadWithTargetMining_80796924772904
// MI455X (gfx1250) — compile-verified
//
#include <hip/hip_runtime.h>

typedef __attribute__((ext_vector_type(2))) float v2f;
typedef __attribute__((ext_vector_type(8))) float v8f;

#define C_CLASSES 100000
#define DIMD      512
#define BATCH     1024
#define KKEEP     50000
#define EPSF      1.1920929e-07f

// ---------------- workspace layout (bytes, all 16B aligned) ----------------
#define OFF_TN      0u          // 1024*512*4      = 2097152
#define OFF_RNORM   2097152u    // 100000*4        = 400000
#define OFF_SCORES  2497152u    // 100000*4        = 400000
#define OFF_KEYS    2897152u    // 100000*4        = 400000
#define OFF_HIST    3297152u    // 256*4           = 1024
#define OFF_SEL     3298176u    // 16*4            = 64
#define OFF_BGT     3298240u    // 512*4           = 2048
#define OFF_BEQ     3300288u
#define OFF_BASEGT  3302336u
#define OFF_BASEEQ  3304384u
#define OFF_POS     3306432u    // 100000*4
#define OFF_SUB     3706432u    // 50000*4

// dynamic LDS for the two GEMM kernels:
//   sW: 64 classes x 512 f32 = 131072 B
//   sRn: 64 f32              = 256 B
//   sT: 16 rows  x 512 f32   = 32768 B
#define GEMM_LDS_BYTES (64 * DIMD * 4 + 256 + 16 * DIMD * 4)

// ---------------- normalize tensor rows (f32) ----------------
__global__ void k_normalize_tensor(const float* __restrict__ t, float* __restrict__ tn) {
    int row = blockIdx.x;
    int tid = threadIdx.x;
    const float* r = t + (size_t)row * DIMD;
    float a = r[tid], b = r[tid + 256];
    float s = a * a + b * b;
    __shared__ float red[8];
    int lane = tid & 31, w = tid >> 5;
#pragma unroll
    for (int off = 16; off; off >>= 1) s += __shfl_xor(s, off);
    if (lane == 0) red[w] = s;
    __syncthreads();
    if (tid == 0) {
        float tt = 0.f;
        for (int i = 0; i < 8; ++i) tt += red[i];
        red[0] = 1.0f / fmaxf(sqrtf(tt), EPSF);
    }
    __syncthreads();
    float sc = red[0];
    float* o = tn + (size_t)row * DIMD;
    o[tid] = a * sc;
    o[tid + 256] = b * sc;
}

// ---------------- mining GEMM: scores[c] = max_b(t_n[b] . w[c]) / ||w[c]|| ----------------
// 128 threads = 4 waves; each wave owns 16 classes; WG owns 64 classes.
__global__ void k_mining(const float* __restrict__ W, const float* __restrict__ tn,
                         float* __restrict__ scores, float* __restrict__ rnorm) {
    extern __shared__ char smem[];
    float* sW  = (float*)smem;                         // [64][512]
    float* sRn = (float*)(smem + 64 * DIMD * 4);       // [64]
    float* sT  = (float*)(smem + 64 * DIMD * 4 + 256); // [16][512]

    int tid = threadIdx.x, lane = tid & 31, w = tid >> 5;
    int blkBase = blockIdx.x * 64;

    // stage 16 weight rows per wave into LDS; compute 1/||w|| per row
    for (int i = 0; i < 16; ++i) {
        int r = w * 16 + i;
        int c = blkBase + r;
        int cc = c < C_CLASSES ? c : (C_CLASSES - 1);
        const float4* wr = (const float4*)(W + (size_t)cc * DIMD);
        float ss = 0.f;
#pragma unroll
        for (int q = 0; q < 4; ++q) {
            float4 f = wr[lane + 32 * q];
            ss += f.x * f.x + f.y * f.y + f.z * f.z + f.w * f.w;
            *(float4*)&sW[r * DIMD + (lane + 32 * q) * 4] = f;
        }
#pragma unroll
        for (int off = 16; off; off >>= 1) ss += __shfl_xor(ss, off);
        if (lane == 0) {
            float rn = 1.0f / fmaxf(sqrtf(ss), EPSF);
            sRn[r] = rn;
            if (c < C_CLASSES) rnorm[c] = rn;
        }
    }
    __syncthreads();

    int cb = w * 16;
    int n = lane & 15, hi = lane >> 4;
    float vmax = -3.0e38f;

    for (int bt = 0; bt < 64; ++bt) {
        {   // stage 16x512 tensor tile (32 KB) cooperatively
            const uint4* src = (const uint4*)(tn + (size_t)bt * 16 * DIMD);
            uint4* dst = (uint4*)sT;
#pragma unroll
            for (int q = 0; q < 16; ++q) dst[tid + 128 * q] = src[tid + 128 * q];
        }
        __syncthreads();
        v8f acc = {};
#pragma unroll 8
        for (int kc = 0; kc < 128; ++kc) {
            // A: lane = M (batch row), K = kc*4 + hi*2 + {0,1}
            v2f a = *(const v2f*)(sT + n * DIMD + kc * 4 + hi * 2);
            // B: lane = N (class col), same K packing
            v2f b = *(const v2f*)(sW + (cb + n) * DIMD + kc * 4 + hi * 2);
            acc = __builtin_amdgcn_wmma_f32_16x16x4_f32(false, a, false, b,
                                                        (short)0, acc, false, false);
        }
        float m = acc[0];
#pragma unroll
        for (int j = 1; j < 8; ++j) m = fmaxf(m, acc[j]);
        vmax = fmaxf(vmax, m);
        __syncthreads();
    }
    // lanes l and l^16 hold the two 8-row halves of the same class column
    float o = __shfl_xor(vmax, 16);
    vmax = fmaxf(vmax, o);
    if (lane < 16) {
        int c = blkBase + cb + lane;
        if (c < C_CLASSES) scores[c] = vmax * sRn[cb + lane];
    }
}

// ---------------- scores[target] = 2.0 ----------------
__global__ void k_set_targets(const int* __restrict__ tgt, float* __restrict__ scores) {
    int i = blockIdx.x * blockDim.x + threadIdx.x;
    if (i < BATCH) scores[tgt[i]] = 2.0f;
}

// ---------------- order-preserving float -> uint key ----------------
__global__ void k_keys(const float* __restrict__ scores, unsigned* __restrict__ keys) {
    int i = blockIdx.x * 256 + threadIdx.x;
    if (i < C_CLASSES) {
        unsigned u = __float_as_uint(scores[i]);
        keys[i] = (u & 0x80000000u) ? ~u : (u | 0x80000000u);
    }
}

__global__ void k_init_sel(unsigned* __restrict__ hist, unsigned* __restrict__ sel) {
    hist[threadIdx.x] = 0u;
    if (threadIdx.x == 0) { sel[0] = 0u; sel[1] = (unsigned)KKEEP; }
}

// ---------------- radix-select histogram pass ----------------
__global__ void k_hist(const unsigned* __restrict__ keys, unsigned* __restrict__ hist,
                       const unsigned* __restrict__ sel, int pass) {
    __shared__ unsigned hl[256];
    int tid = threadIdx.x;
    hl[tid] = 0u;
    __syncthreads();
    int i = blockIdx.x * 256 + tid;
    if (i < C_CLASSES) {
        unsigned u = keys[i];
        int shift = 24 - 8 * pass;
        bool ok = (pass == 0) || ((u >> (shift + 8)) == sel[0]);
        if (ok) atomicAdd(&hl[(u >> shift) & 255u], 1u);
    }
    __syncthreads();
    if (hl[tid]) atomicAdd(&hist[tid], hl[tid]);
}

// pick byte containing the r-th largest; update (prefix, r); zero hist for next pass
__global__ void k_select(unsigned* __restrict__ hist, unsigned* __restrict__ sel) {
    unsigned r = sel[1], cum = 0;
    for (int b = 255; b >= 0; --b) {
        unsigned h = hist[b];
        if (cum + h >= r) { sel[0] = (sel[0] << 8) | (unsigned)b; sel[1] = r - cum; break; }
        cum += h;
    }
    for (int i = 0; i < 256; ++i) hist[i] = 0u;
}

// ---------------- per-block counts of (>tau) and (==tau) ----------------
__global__ void k_count(const unsigned* __restrict__ keys, const unsigned* __restrict__ sel,
                        unsigned* __restrict__ bgt, unsigned* __restrict__ beq) {
    int tid = threadIdx.x, lane = tid & 31, w = tid >> 5;
    int i = blockIdx.x * 256 + tid;
    unsigned tau = sel[0];
    bool gt = false, eq = false;
    if (i < C_CLASSES) { unsigned u = keys[i]; gt = u > tau; eq = u == tau; }
    unsigned mg = (unsigned)__ballot(gt);
    unsigned me = (unsigned)__ballot(eq);
    __shared__ unsigned wg[8], we[8];
    if (lane == 0) { wg[w] = __popc(mg); we[w] = __popc(me); }
    __syncthreads();
    if (tid == 0) {
        unsigned g = 0, e = 0;
        for (int k = 0; k < 8; ++k) { g += wg[k]; e += we[k]; }
        bgt[blockIdx.x] = g; beq[blockIdx.x] = e;
    }
}

// ---------------- exclusive scan of block counts (single block, 512 threads) ----------------
__global__ void k_scan(const unsigned* __restrict__ bgt, const unsigned* __restrict__ beq,
                       unsigned* __restrict__ basegt, unsigned* __restrict__ baseeq, int nb) {
    __shared__ unsigned sg[512], se_[512];
    int t = threadIdx.x;
    unsigned og = t < nb ? bgt[t] : 0u;
    unsigned oe = t < nb ? beq[t] : 0u;
    sg[t] = og; se_[t] = oe;
    __syncthreads();
    for (int d = 1; d < 512; d <<= 1) {
        unsigned ag = (t >= d) ? sg[t - d] : 0u;
        unsigned ae = (t >= d) ? se_[t - d] : 0u;
        __syncthreads();
        sg[t] += ag; se_[t] += ae;
        __syncthreads();
    }
    if (t < nb) { basegt[t] = sg[t] - og; baseeq[t] = se_[t] - oe; }
}

// ---------------- compact selected indices in ascending order ----------------
__global__ void k_compact(const unsigned* __restrict__ keys, const unsigned* __restrict__ sel,
                          const unsigned* __restrict__ basegt, const unsigned* __restrict__ baseeq,
                          int* __restrict__ sub, int* __restrict__ posOf) {
    int tid = threadIdx.x, lane = tid & 31, w = tid >> 5;
    int i = blockIdx.x * 256 + tid;
    unsigned tau = sel[0], r = sel[1];
    bool gt = false, eq = false;
    if (i < C_CLASSES) { unsigned u = keys[i]; gt = u > tau; eq = u == tau; }
    unsigned mg = (unsigned)__ballot(gt);
    unsigned me = (unsigned)__ballot(eq);
    __shared__ unsigned wg[8], we[8];
    if (lane == 0) { wg[w] = __popc(mg); we[w] = __popc(me); }
    __syncthreads();
    unsigned lmask = (1u << lane) - 1u;
    unsigned lgt = __popc(mg & lmask);
    unsigned leq = __popc(me & lmask);
    unsigned bg = 0, be = 0;
    for (int k = 0; k < w; ++k) { bg += wg[k]; be += we[k]; }
    unsigned gpref = basegt[blockIdx.x] + bg + lgt;  // # selected-GT with index < i
    unsigned epref = baseeq[blockIdx.x] + be + leq;  // # EQ with index < i
    if (gt) {
        unsigned pos = gpref + (epref < r ? epref : r);
        sub[pos] = i; posOf[i] = (int)pos;
    } else if (eq && epref < r) {
        unsigned pos = gpref + epref;
        sub[pos] = i; posOf[i] = (int)pos;
    }
}

// ---------------- new_target = position of target class in subsample ----------------
__global__ void k_newtarget(const int* __restrict__ tgt, const int* __restrict__ posOf,
                            float* __restrict__ outTail) {
    int i = blockIdx.x * 256 + threadIdx.x;
    if (i < BATCH) outTail[i] = (float)posOf[tgt[i]];
}

// ---------------- logits GEMM over gathered rows ----------------
__global__ void k_logits(const float* __restrict__ W, const float* __restrict__ tn,
                         const int* __restrict__ sub, const float* __restrict__ rnorm,
                         float* __restrict__ out) {
    extern __shared__ char smem[];
    float* sW  = (float*)smem;
    float* sRn = (float*)(smem + 64 * DIMD * 4);
    float* sT  = (float*)(smem + 64 * DIMD * 4 + 256);

    int tid = threadIdx.x, lane = tid & 31, w = tid >> 5;
    int kBase = blockIdx.x * 64;

    for (int i = 0; i < 16; ++i) {
        int r = w * 16 + i;
        int col = kBase + r;
        int cls = col < KKEEP ? sub[col] : 0;
        const float4* wr = (const float4*)(W + (size_t)cls * DIMD);
#pragma unroll
        for (int q = 0; q < 4; ++q)
            *(float4*)&sW[r * DIMD + (lane + 32 * q) * 4] = wr[lane + 32 * q];
        if (lane == 0) sRn[r] = rnorm[cls];
    }
    __syncthreads();

    int cb = w * 16;
    int n = lane & 15, hi = lane >> 4;
    int col = kBase + cb + n;

    for (int bt = 0; bt < 64; ++bt) {
        {
            const uint4* src = (const uint4*)(tn + (size_t)bt * 16 * DIMD);
            uint4* dst = (uint4*)sT;
#pragma unroll
            for (int q = 0; q < 16; ++q) dst[tid + 128 * q] = src[tid + 128 * q];
        }
        __syncthreads();
        v8f acc = {};
#pragma unroll 8
        for (int kc = 0; kc < 128; ++kc) {
            v2f a = *(const v2f*)(sT + n * DIMD + kc * 4 + hi * 2);
            v2f b = *(const v2f*)(sW + (cb + n) * DIMD + kc * 4 + hi * 2);
            acc = __builtin_amdgcn_wmma_f32_16x16x4_f32(false, a, false, b,
                                                        (short)0, acc, false, false);
        }
        if (col < KKEEP) {
            float rn = sRn[cb + n];
#pragma unroll
            for (int j = 0; j < 8; ++j) {
                int row = bt * 16 + j + hi * 8;  // D: VGPR j -> M = j (+8 for upper lanes)
                out[(size_t)row * KKEEP + col] = acc[j] * rn;
            }
        }
        __syncthreads();
    }
}

extern "C" void kernel_launch(void* const* d_in, const int* in_sizes, int n_in,
                              void* d_out, int out_size, void* d_ws, size_t ws_size,
                              hipStream_t stream) {
    (void)in_sizes; (void)n_in; (void)out_size; (void)ws_size;
    const float* tensor = (const float*)d_in[0];
    const int*   target = (const int*)d_in[1];
    const float* weight = (const float*)d_in[2];
    float* out = (float*)d_out;
    char* ws = (char*)d_ws;

    float*    tn     = (float*)(ws + OFF_TN);
    float*    rnorm  = (float*)(ws + OFF_RNORM);
    float*    scores = (float*)(ws + OFF_SCORES);
    unsigned* keys   = (unsigned*)(ws + OFF_KEYS);
    unsigned* hist   = (unsigned*)(ws + OFF_HIST);
    unsigned* sel    = (unsigned*)(ws + OFF_SEL);
    unsigned* bgt    = (unsigned*)(ws + OFF_BGT);
    unsigned* beq    = (unsigned*)(ws + OFF_BEQ);
    unsigned* basegt = (unsigned*)(ws + OFF_BASEGT);
    unsigned* baseeq = (unsigned*)(ws + OFF_BASEEQ);
    int*      posOf  = (int*)(ws + OFF_POS);
    int*      sub    = (int*)(ws + OFF_SUB);

    const int nbElem = (C_CLASSES + 255) / 256;          // 391
    const int nbMine = (C_CLASSES + 63) / 64;            // 1563
    const int nbLog  = (KKEEP + 63) / 64;                // 782

    k_normalize_tensor<<<BATCH, 256, 0, stream>>>(tensor, tn);
    k_mining<<<nbMine, 128, GEMM_LDS_BYTES, stream>>>(weight, tn, scores, rnorm);
    k_set_targets<<<4, 256, 0, stream>>>(target, scores);
    k_keys<<<nbElem, 256, 0, stream>>>(scores, keys);
    k_init_sel<<<1, 256, 0, stream>>>(hist, sel);
    for (int p = 0; p < 4; ++p) {
        k_hist<<<nbElem, 256, 0, stream>>>(keys, hist, sel, p);
        k_select<<<1, 1, 0, stream>>>(hist, sel);
    }
    k_count<<<nbElem, 256, 0, stream>>>(keys, sel, bgt, beq);
    k_scan<<<1, 512, 0, stream>>>(bgt, beq, basegt, baseeq, nbElem);
    k_compact<<<nbElem, 256, 0, stream>>>(keys, sel, basegt, baseeq, sub, posOf);
    k_newtarget<<<4, 256, 0, stream>>>(target, posOf, out + (size_t)BATCH * KKEEP);
    k_logits<<<nbLog, 128, GEMM_LDS_BYTES, stream>>>(weight, tn, sub, rnorm, out);
}